// ScatterAndSoftmaxLayer_86028194939133
// MI455X (gfx1250) — compile-verified
//
#include <hip/hip_runtime.h>
#include <float.h>

#define WAVE 32
#define CAP  128          // staged edges per node (8KB LDS per wave)
#define WPB  8            // waves per block

#define AS1 __attribute__((address_space(1)))
#define AS3 __attribute__((address_space(3)))

typedef __attribute__((ext_vector_type(4))) int v4i;

__device__ __forceinline__ float4 fmax4(float4 a, float4 b) {
    return make_float4(fmaxf(a.x, b.x), fmaxf(a.y, b.y),
                       fmaxf(a.z, b.z), fmaxf(a.w, b.w));
}

// One 16B async copy global->LDS (ASYNCcnt-tracked, no VGPR round-trip).
// Builtin signature (from hipcc diagnostic): arg0 = v4i addrspace(1)*,
// then LDS dest, imm offset, imm cpol.
__device__ __forceinline__ void async_copy16(const char* g, char* l) {
#if defined(__has_builtin) && __has_builtin(__builtin_amdgcn_global_load_async_to_lds_b128)
    __builtin_amdgcn_global_load_async_to_lds_b128(
        (AS1 v4i*)(size_t)g,   // generic==global address numerically on AMDGCN
        (AS3 v4i*)l,           // addrspacecast generic -> local
        0, 0);
#else
    unsigned     loff = (unsigned)(size_t)(AS3 char*)l;   // raw LDS byte offset
    unsigned long long ga = (unsigned long long)(size_t)g;
    asm volatile("global_load_async_to_lds_b128 %0, %1, off"
                 :: "v"(loff), "v"(ga) : "memory");
#endif
}

__device__ __forceinline__ void wait_async0() {
#if defined(__has_builtin) && __has_builtin(__builtin_amdgcn_s_wait_asynccnt)
    __builtin_amdgcn_s_wait_asynccnt(0);
#else
    asm volatile("s_wait_asynccnt 0" ::: "memory");
#endif
    asm volatile("" ::: "memory");  // keep LDS reads below the wait
}

// Kernel 1: CSR-style segment offsets via binary search on the sorted id array.
__global__ void seg_bounds_kernel(const int* __restrict__ ids,
                                  int* __restrict__ node_off, int N, int E) {
    int n = blockIdx.x * blockDim.x + threadIdx.x;
    if (n > N) return;
    int lo = 0, hi = E;
    while (lo < hi) {
        int mid = (lo + hi) >> 1;
        if (ids[mid] < n) lo = mid + 1; else hi = mid;
    }
    node_off[n] = lo;
}

// Kernel 2: fused segment softmax, one wave32 per node.
// Stage the node's edge span into LDS via global_load_async_to_lds_b128
// (one cold HBM read), then do max / exp-sum / normalize entirely from LDS.
// Lane layout: fq = lane&3 (feature quad), sub = lane>>2 (edge-in-group-of-8);
// every wave iteration touches 512B contiguous (bank-conflict-free).
// Deterministic: fixed reduction order, no atomics. Stats live in registers.
__global__ void __launch_bounds__(WAVE * WPB)
seg_softmax_kernel(const float* __restrict__ Eset,
                   const int* __restrict__ node_off,
                   float* __restrict__ out, int N) {
    __shared__ __align__(16) char smem[WPB][CAP * 64];

    int w    = threadIdx.x >> 5;
    int node = blockIdx.x * WPB + w;
    if (node >= N) return;
    int lane = threadIdx.x & (WAVE - 1);
    int sub  = lane >> 2;
    int fq   = lane & 3;
    long lo = node_off[node];
    long hi = node_off[node + 1];
    if (lo >= hi) return;                 // empty segment: never referenced
    int nE = (int)(hi - lo);

    float4* __restrict__ O4 = (float4*)out;
    float4 m = make_float4(-FLT_MAX, -FLT_MAX, -FLT_MAX, -FLT_MAX);
    float4 s = make_float4(0.f, 0.f, 0.f, 0.f);

    if (nE <= CAP) {
        // ---- Stage span to LDS: nE edges * 64B, 512B per wave instruction.
        const char* gbase = (const char*)Eset + (size_t)lo * 64;
        char*       lbase = smem[w];
        int units = nE * 4;                       // 16B units
        for (int u = lane; u < units; u += WAVE)
            async_copy16(gbase + (size_t)u * 16, lbase + (size_t)u * 16);
        wait_async0();

        const float4* L4 = (const float4*)lbase;
        // Pass 1: segment max (LDS)
        for (int i = sub; i < nE; i += 8) m = fmax4(m, L4[i * 4 + fq]);
#pragma unroll
        for (int off = 4; off < WAVE; off <<= 1) {
            m.x = fmaxf(m.x, __shfl_xor(m.x, off, WAVE));
            m.y = fmaxf(m.y, __shfl_xor(m.y, off, WAVE));
            m.z = fmaxf(m.z, __shfl_xor(m.z, off, WAVE));
            m.w = fmaxf(m.w, __shfl_xor(m.w, off, WAVE));
        }
        // Pass 2: segment sum of exp(v - m) (LDS)
        for (int i = sub; i < nE; i += 8) {
            float4 v = L4[i * 4 + fq];
            s.x += __expf(v.x - m.x);
            s.y += __expf(v.y - m.y);
            s.z += __expf(v.z - m.z);
            s.w += __expf(v.w - m.w);
        }
#pragma unroll
        for (int off = 4; off < WAVE; off <<= 1) {
            s.x += __shfl_xor(s.x, off, WAVE);
            s.y += __shfl_xor(s.y, off, WAVE);
            s.z += __shfl_xor(s.z, off, WAVE);
            s.w += __shfl_xor(s.w, off, WAVE);
        }
        float4 inv = make_float4(1.f / s.x, 1.f / s.y, 1.f / s.z, 1.f / s.w);
        // Pass 3: normalized output (LDS -> global, 512B contiguous stores)
        for (int i = sub; i < nE; i += 8) {
            float4 v = L4[i * 4 + fq];
            float4 r;
            r.x = __expf(v.x - m.x) * inv.x;
            r.y = __expf(v.y - m.y) * inv.y;
            r.z = __expf(v.z - m.z) * inv.z;
            r.w = __expf(v.w - m.w) * inv.w;
            O4[(lo + i) * 4 + fq] = r;
        }
    } else {
        // ---- Fallback for pathological high-degree nodes: global-read path.
        const float4* __restrict__ E4 = (const float4*)Eset;
        for (long e = lo + sub; e < hi; e += 8) {
            __builtin_prefetch(&E4[(e + 8) * 4 + fq], 0, 0);
            m = fmax4(m, E4[e * 4 + fq]);
        }
#pragma unroll
        for (int off = 4; off < WAVE; off <<= 1) {
            m.x = fmaxf(m.x, __shfl_xor(m.x, off, WAVE));
            m.y = fmaxf(m.y, __shfl_xor(m.y, off, WAVE));
            m.z = fmaxf(m.z, __shfl_xor(m.z, off, WAVE));
            m.w = fmaxf(m.w, __shfl_xor(m.w, off, WAVE));
        }
        for (long e = lo + sub; e < hi; e += 8) {
            float4 v = E4[e * 4 + fq];
            s.x += __expf(v.x - m.x);
            s.y += __expf(v.y - m.y);
            s.z += __expf(v.z - m.z);
            s.w += __expf(v.w - m.w);
        }
#pragma unroll
        for (int off = 4; off < WAVE; off <<= 1) {
            s.x += __shfl_xor(s.x, off, WAVE);
            s.y += __shfl_xor(s.y, off, WAVE);
            s.z += __shfl_xor(s.z, off, WAVE);
            s.w += __shfl_xor(s.w, off, WAVE);
        }
        float4 inv = make_float4(1.f / s.x, 1.f / s.y, 1.f / s.z, 1.f / s.w);
        for (long e = lo + sub; e < hi; e += 8) {
            float4 v = E4[e * 4 + fq];
            float4 r;
            r.x = __expf(v.x - m.x) * inv.x;
            r.y = __expf(v.y - m.y) * inv.y;
            r.z = __expf(v.z - m.z) * inv.z;
            r.w = __expf(v.w - m.w) * inv.w;
            O4[e * 4 + fq] = r;
        }
    }
}

extern "C" void kernel_launch(void* const* d_in, const int* in_sizes, int n_in,
                              void* d_out, int out_size, void* d_ws, size_t ws_size,
                              hipStream_t stream) {
    // Inputs: V_set [1,N,16] f32, E_set [1,E,16] f32, node_ids [1,E] i32
    const int N = in_sizes[0] / 16;
    const int E = in_sizes[2];
    const float* Eset = (const float*)d_in[1];
    const int*   ids  = (const int*)d_in[2];
    float* out = (float*)d_out;

    int* node_off = (int*)d_ws;   // only workspace: node_off[N+1]

    {   // segment boundaries
        int threads = 256;
        int blocks  = (N + 1 + threads - 1) / threads;
        seg_bounds_kernel<<<blocks, threads, 0, stream>>>(ids, node_off, N, E);
    }
    {   // fused softmax: one wave per node, 8 waves (64KB LDS) per block
        int threads = WAVE * WPB;
        int blocks  = (N + WPB - 1) / WPB;
        seg_softmax_kernel<<<blocks, threads, 0, stream>>>(Eset, node_off, out, N);
    }
}